// NewtonLayer_70102456205500
// MI455X (gfx1250) — compile-verified
//
#include <hip/hip_runtime.h>

// CDNA5 / gfx1250. wave32. WMMA f32 16x16x4 for the A*sin(y) coupling GEMM;
// cooperative 16-lane Gaussian elimination (register-resident, fully unrolled)
// for the per-system 16x16 Newton solve.

typedef __attribute__((ext_vector_type(2))) float v2f; // 16x4 / 4x16 f32 WMMA operand (2 VGPRs)
typedef __attribute__((ext_vector_type(8))) float v8f; // 16x16 f32 accumulator (8 VGPRs)

#define NVARS      16
#define SYS_PER_WG 16
#define WG_THREADS (SYS_PER_WG * NVARS)   // 256 threads = 8 waves
#define MAX_ITER   20
#define STEP       0.1f

// v_rcp_f32 + one Newton-Raphson refinement: ~1 ulp, far cheaper than full divide.
__device__ __forceinline__ float fast_rcp(float a) {
    float r = __builtin_amdgcn_rcpf(a);
    r = r * (2.0f - a * r);
    return r;
}

__global__ __launch_bounds__(WG_THREADS)
void newton16_kernel(const float* __restrict__ y_in,
                     const float* __restrict__ x_in,
                     const float* __restrict__ A,
                     float* __restrict__ y_out,
                     int batch)
{
    // Padded to 17 floats/row: 17 coprime with 64 banks -> conflict-free column reads.
    __shared__ float sA  [NVARS][NVARS + 1];      // coupling matrix
    __shared__ float sSin[NVARS][NVARS + 1];      // [var j][system s] : GEMM B operand
    __shared__ float sCos[SYS_PER_WG][NVARS];     // [system s][var j] : Jacobian build
    __shared__ float sR  [NVARS][NVARS + 1];      // [var i][system s] : GEMM result

    const int t = threadIdx.x;
    const int s = t >> 4;          // local system (0..15)
    const int i = t & 15;          // row / variable index (0..15)
    const int gsys   = blockIdx.x * SYS_PER_WG + s;
    const int gclamp = (gsys < batch) ? gsys : (batch - 1);

    // Stage A into LDS once (256 threads <-> 256 elements).
    sA[t >> 4][t & 15] = A[t];
    __syncthreads();

    // Each lane keeps its Jacobian-template row of A in registers.
    float Arow[NVARS];
    #pragma unroll
    for (int j = 0; j < NVARS; ++j) Arow[j] = sA[i][j];

    // Wave 0 preloads the constant WMMA A-fragments of the coupling matrix.
    // 16x4 f32 A-operand layout: lanes 0-15 hold K = kb, kb+1 ; lanes 16-31 hold K = kb+2, kb+3.
    v2f afrag[4] = {};
    if (t < 32) {
        const int m  = t & 15;
        const int lh = t >> 4;                       // lane-half select
        #pragma unroll
        for (int kc = 0; kc < 4; ++kc) {
            const int kb = 4 * kc + 2 * lh;
            afrag[kc].x = sA[m][kb];
            afrag[kc].y = sA[m][kb + 1];
        }
    }

    float y  = y_in[gclamp * NVARS + i];
    float xv = x_in[gclamp * NVARS + i];

    for (int it = 0; it < MAX_ITER; ++it) {
        float sy, cy;
        __sincosf(y, &sy, &cy);

        __syncthreads();                     // previous iteration's LDS reads complete
        sSin[i][s] = sy;
        sCos[s][i] = cy;
        __syncthreads();

        // ---- Coupling GEMM: R[i][s] = sum_j A[i][j] * sin(y[s][j]) via WMMA f32 16x16x4.
        // Branch is wave-uniform (threads 0-31 == wave 0) -> EXEC all-1s inside WMMA.
        if (t < 32) {
            const int n  = t & 15;
            const int lh = t >> 4;
            v8f acc = {};
            #pragma unroll
            for (int kc = 0; kc < 4; ++kc) {
                v2f b;                                   // 4x16 B operand chunk
                b.x = sSin[4 * kc + lh][n];
                b.y = sSin[4 * kc + 2 + lh][n];
                acc = __builtin_amdgcn_wmma_f32_16x16x4_f32(
                        false, afrag[kc], false, b, (short)0, acc, false, false);
            }
            // D layout: VGPR v, lanes 0-15 -> M=v, lanes 16-31 -> M=v+8, N=lane%16.
            #pragma unroll
            for (int v = 0; v < 8; ++v)
                sR[v + 8 * lh][n] = acc[v];
        }
        __syncthreads();

        // ---- Residual and Jacobian row (register-resident).
        const float r    = y * y * y + sR[i][s] - xv;
        const float diag = 3.0f * y * y;
        float Jr[NVARS];
        #pragma unroll
        for (int j = 0; j < NVARS; ++j) {
            const float v = Arow[j] * sCos[s][j];
            Jr[j] = (j == i) ? (v + diag) : v;   // compile-time j vs runtime i -> v_cndmask
        }

        // ---- Forward elimination (16 lanes cooperate; pivot row broadcast via 16-wide shfl).
        // All shuffles execute in uniform control flow; f==0 masks rows i<=k.
        float rhs = -r;
        #pragma unroll
        for (int k = 0; k < NVARS - 1; ++k) {
            const float pk   = __shfl(Jr[k], k, 16);
            const float prhs = __shfl(rhs,  k, 16);
            const float f    = (i > k) ? Jr[k] * fast_rcp(pk) : 0.0f;
            rhs -= f * prhs;
            #pragma unroll
            for (int j = k + 1; j < NVARS; ++j) {
                const float pj = __shfl(Jr[j], k, 16);
                Jr[j] -= f * pj;
            }
        }

        // ---- Back substitution.
        float delta = 0.0f;
        #pragma unroll
        for (int k = NVARS - 1; k >= 0; --k) {
            const float num = __shfl(rhs,   k, 16);
            const float den = __shfl(Jr[k], k, 16);
            const float xk  = num * fast_rcp(den);
            if (i == k) delta = xk;
            rhs -= (i < k) ? Jr[k] * xk : 0.0f;
        }

        y += STEP * delta;
    }

    if (gsys < batch) y_out[gsys * NVARS + i] = y;
}

extern "C" void kernel_launch(void* const* d_in, const int* in_sizes, int n_in,
                              void* d_out, int out_size, void* d_ws, size_t ws_size,
                              hipStream_t stream) {
    const float* y = (const float*)d_in[0];
    const float* x = (const float*)d_in[1];
    const float* A = (const float*)d_in[2];
    float* out = (float*)d_out;

    const int batch  = in_sizes[0] / NVARS;           // 32768
    const int blocks = (batch + SYS_PER_WG - 1) / SYS_PER_WG;

    newton16_kernel<<<blocks, WG_THREADS, 0, stream>>>(y, x, A, out, batch);
}